// ClusterMIcc_54477365182874
// MI455X (gfx1250) — compile-verified
//
#include <hip/hip_runtime.h>
#include <hip/hip_bf16.h>

typedef __attribute__((ext_vector_type(2))) float v2f;
typedef __attribute__((ext_vector_type(8))) float v8f;

#define NPTS 8192
#define DX 64
#define DY 16
#define TK 6   // K+1 neighbors including self

// ---------------------------------------------------------------- digamma
__device__ __forceinline__ float digammaf(float x) {
  float r = 0.f;
  while (x < 6.f) { r -= 1.f / x; x += 1.f; }
  float inv = 1.f / x, inv2 = inv * inv;
  float res = __logf(x) - 0.5f * inv
            - inv2 * (1.f / 12.f - inv2 * (1.f / 120.f - inv2 * (1.f / 252.f)));
  return res + r;
}

// ---------------------------------------------------------------- kernel 1: row norms (+ zero accumulator)
__global__ __launch_bounds__(256) void k_norms(const float* __restrict__ X,
                                               const float* __restrict__ Y,
                                               float* __restrict__ sqx,
                                               float* __restrict__ sqy,
                                               float* __restrict__ accum) {
  int i = blockIdx.x * blockDim.x + threadIdx.x;
  if (i == 0) *accum = 0.f;
  if (i < NPTS) {
    const float* xr = X + (size_t)i * DX;
    float s = 0.f;
#pragma unroll
    for (int k = 0; k < DX; ++k) s += xr[k] * xr[k];
    sqx[i] = s;
    const float* yr = Y + (size_t)i * DY;
    float t = 0.f;
#pragma unroll
    for (int k = 0; k < DY; ++k) t += yr[k] * yr[k];
    sqy[i] = t;
  }
}

// ---------------------------------------------------------------- kernel 2: per-row 6th-smallest of max(d2x,d2y)
__global__ __launch_bounds__(256) void k_select(const float* __restrict__ X,
                                                const float* __restrict__ Y,
                                                const float* __restrict__ sqx,
                                                const float* __restrict__ sqy,
                                                int* __restrict__ kth) {
  __shared__ float lXi[16][DX + 4];
  __shared__ float lYi[16][DY + 4];
  __shared__ float lsqx[16];
  __shared__ float lsqy[16];
  __shared__ float ltile[8][16][17];   // per-wave private tile buffer
  __shared__ float mv[16][16 * TK];
  __shared__ int   mid[16][16 * TK];

  const int tid  = threadIdx.x;
  const int w    = tid >> 5;
  const int lane = tid & 31;
  const int hi   = lane >> 4;
  const int n    = lane & 15;
  const int ibase = blockIdx.x * 16;

  for (int idx = tid; idx < 16 * DX; idx += 256) {
    int r = idx >> 6, k = idx & 63;
    lXi[r][k] = X[(size_t)(ibase + r) * DX + k];
  }
  {
    int r = tid >> 4, k = tid & 15;   // 256 threads == 16*16 elems
    lYi[r][k] = Y[(size_t)(ibase + r) * DY + k];
  }
  if (tid < 16) { lsqx[tid] = sqx[ibase + tid]; lsqy[tid] = sqy[ibase + tid]; }
  __syncthreads();

  float tv[TK]; int ti[TK];
#pragma unroll
  for (int s = 0; s < TK; ++s) { tv[s] = 3.4e38f; ti[s] = 0; }

  const int myrow = lane >> 1;
  const int cbase = (lane & 1) * 8;

  // Each wave sweeps its own tiles into its own LDS buffer: no block barrier
  // needed inside the loop (per-wave LDS ops are in-order; compiler inserts
  // the s_wait_dscnt for the aliasing store->load).
  for (int t = w; t < NPTS / 16; t += 8) {
    const int jbase = t * 16;

    v8f ax = {};
    const float* Bx = X + (size_t)jbase * DX;
#pragma unroll
    for (int k0 = 0; k0 < DX; k0 += 4) {
      const int kk = k0 + 2 * hi;
      v2f a; a[0] = lXi[n][kk]; a[1] = lXi[n][kk + 1];
      v2f b = *(const v2f*)(Bx + (size_t)n * DX + kk);
      ax = __builtin_amdgcn_wmma_f32_16x16x4_f32(false, a, false, b, (short)0, ax, false, false);
    }
    v8f ay = {};
    const float* By = Y + (size_t)jbase * DY;
#pragma unroll
    for (int k0 = 0; k0 < DY; k0 += 4) {
      const int kk = k0 + 2 * hi;
      v2f a; a[0] = lYi[n][kk]; a[1] = lYi[n][kk + 1];
      v2f b = *(const v2f*)(By + (size_t)n * DY + kk);
      ay = __builtin_amdgcn_wmma_f32_16x16x4_f32(false, a, false, b, (short)0, ay, false, false);
    }

    const float sxj = sqx[jbase + n];
    const float syj = sqy[jbase + n];
#pragma unroll
    for (int r = 0; r < 8; ++r) {
      const int m = r + 8 * hi;
      float d2x = fmaxf(lsqx[m] + sxj - 2.f * ax[r], 0.f);
      float d2y = fmaxf(lsqy[m] + syj - 2.f * ay[r], 0.f);
      ltile[w][m][n] = fmaxf(d2x, d2y);
    }
    __builtin_amdgcn_wave_barrier();   // keep store phase before load phase

#pragma unroll
    for (int c = 0; c < 8; ++c) {
      float v = ltile[w][myrow][cbase + c];
      int j = jbase + cbase + c;
      if (v < tv[TK - 1]) {
        tv[TK - 1] = v; ti[TK - 1] = j;
#pragma unroll
        for (int s = TK - 1; s > 0; --s) {
          if (tv[s] < tv[s - 1]) {
            float fv = tv[s]; tv[s] = tv[s - 1]; tv[s - 1] = fv;
            int iv = ti[s]; ti[s] = ti[s - 1]; ti[s - 1] = iv;
          }
        }
      }
    }
    __builtin_amdgcn_wave_barrier();   // keep loads before next tile's stores
  }

  // merge 16 partial top-6 lists per row
  const int slot = w * 2 + (lane & 1);
#pragma unroll
  for (int s = 0; s < TK; ++s) {
    mv[myrow][slot * TK + s]  = tv[s];
    mid[myrow][slot * TK + s] = ti[s];
  }
  __syncthreads();

  if (tid < 16) {
    float bv[TK]; int bi[TK];
#pragma unroll
    for (int s = 0; s < TK; ++s) { bv[s] = 3.4e38f; bi[s] = 0; }
    for (int c = 0; c < 16 * TK; ++c) {
      float v = mv[tid][c]; int j = mid[tid][c];
      if (v < bv[TK - 1]) {
        bv[TK - 1] = v; bi[TK - 1] = j;
#pragma unroll
        for (int s = TK - 1; s > 0; --s) {
          if (bv[s] < bv[s - 1]) {
            float fv = bv[s]; bv[s] = bv[s - 1]; bv[s - 1] = fv;
            int iv = bi[s]; bi[s] = bi[s - 1]; bi[s - 1] = iv;
          }
        }
      }
    }
    kth[ibase + tid] = bi[TK - 1];
  }
}

// ---------------------------------------------------------------- kernel 3: neighbor counts + digamma accumulation
__global__ __launch_bounds__(256) void k_count(const float* __restrict__ X,
                                               const float* __restrict__ Y,
                                               const float* __restrict__ sqx,
                                               const float* __restrict__ sqy,
                                               const int* __restrict__ kth,
                                               float* __restrict__ accum) {
  __shared__ float lXi[16][DX + 4];
  __shared__ float lXk[16][DX + 4];
  __shared__ float lYi[16][DY + 4];
  __shared__ float lYk[16][DY + 4];
  __shared__ float lsqxi[16], lsqxk[16], lsqyi[16], lsqyk[16];
  __shared__ int   lkth[16];
  __shared__ unsigned cntx[16], cnty[16];
  __shared__ float red[16];

  const int tid  = threadIdx.x;
  const int w    = tid >> 5;
  const int lane = tid & 31;
  const int hi   = lane >> 4;
  const int n    = lane & 15;
  const int ibase = blockIdx.x * 16;

  if (tid < 16) {
    int kr = kth[ibase + tid];
    lkth[tid]  = kr;
    lsqxi[tid] = sqx[ibase + tid]; lsqxk[tid] = sqx[kr];
    lsqyi[tid] = sqy[ibase + tid]; lsqyk[tid] = sqy[kr];
    cntx[tid] = 0u; cnty[tid] = 0u;
  }
  __syncthreads();

  for (int idx = tid; idx < 16 * DX; idx += 256) {
    int r = idx >> 6, k = idx & 63;
    lXi[r][k] = X[(size_t)(ibase + r) * DX + k];
    lXk[r][k] = X[(size_t)lkth[r] * DX + k];
  }
  {
    int r = tid >> 4, k = tid & 15;
    lYi[r][k] = Y[(size_t)(ibase + r) * DY + k];
    lYk[r][k] = Y[(size_t)lkth[r] * DY + k];
  }
  __syncthreads();

  int cx[8], cy[8];
#pragma unroll
  for (int r = 0; r < 8; ++r) { cx[r] = 0; cy[r] = 0; }

  for (int t = w; t < NPTS / 16; t += 8) {
    const int jbase = t * 16;

    v8f axi = {}, axk = {};
    const float* Bx = X + (size_t)jbase * DX;
#pragma unroll
    for (int k0 = 0; k0 < DX; k0 += 4) {
      const int kk = k0 + 2 * hi;
      v2f b = *(const v2f*)(Bx + (size_t)n * DX + kk);
      v2f ai; ai[0] = lXi[n][kk]; ai[1] = lXi[n][kk + 1];
      v2f ak; ak[0] = lXk[n][kk]; ak[1] = lXk[n][kk + 1];
      axi = __builtin_amdgcn_wmma_f32_16x16x4_f32(false, ai, false, b, (short)0, axi, false, false);
      axk = __builtin_amdgcn_wmma_f32_16x16x4_f32(false, ak, false, b, (short)0, axk, false, false);
    }
    v8f ayi = {}, ayk = {};
    const float* By = Y + (size_t)jbase * DY;
#pragma unroll
    for (int k0 = 0; k0 < DY; k0 += 4) {
      const int kk = k0 + 2 * hi;
      v2f b = *(const v2f*)(By + (size_t)n * DY + kk);
      v2f ai; ai[0] = lYi[n][kk]; ai[1] = lYi[n][kk + 1];
      v2f ak; ak[0] = lYk[n][kk]; ak[1] = lYk[n][kk + 1];
      ayi = __builtin_amdgcn_wmma_f32_16x16x4_f32(false, ai, false, b, (short)0, ayi, false, false);
      ayk = __builtin_amdgcn_wmma_f32_16x16x4_f32(false, ak, false, b, (short)0, ayk, false, false);
    }

    const float sxj = sqx[jbase + n];
    const float syj = sqy[jbase + n];
#pragma unroll
    for (int r = 0; r < 8; ++r) {
      const int m = r + 8 * hi;
      float dxi = fmaxf(lsqxi[m] + sxj - 2.f * axi[r], 0.f);
      float dxk = fmaxf(lsqxk[m] + sxj - 2.f * axk[r], 0.f);
      cx[r] += (dxi <= dxk) ? 1 : 0;
      float dyi = fmaxf(lsqyi[m] + syj - 2.f * ayi[r], 0.f);
      float dyk = fmaxf(lsqyk[m] + syj - 2.f * ayk[r], 0.f);
      cy[r] += (dyi <= dyk) ? 1 : 0;
    }
  }

#pragma unroll
  for (int r = 0; r < 8; ++r) {
    const int m = r + 8 * hi;
    atomicAdd(&cntx[m], (unsigned)cx[r]);
    atomicAdd(&cnty[m], (unsigned)cy[r]);
  }
  __syncthreads();

  if (tid < 16) {
    red[tid] = digammaf((float)cntx[tid] + 1.f) + digammaf((float)cnty[tid] + 1.f);
  }
  __syncthreads();
  if (tid == 0) {
    float s = 0.f;
#pragma unroll
    for (int i = 0; i < 16; ++i) s += red[i];
    atomicAdd(accum, s);
  }
}

// ---------------------------------------------------------------- kernel 4: finalize MI scalar
__global__ void k_final(const float* __restrict__ accum, float* __restrict__ out) {
  float n_avg = *accum / (float)NPTS;
  float k_dig = digammaf(5.f) - 0.2f;           // digamma(K) - 1/K, K=5
  float n_dig = digammaf((float)NPTS);
  float mi = (n_dig + k_dig - n_avg) / __logf(2.f);
  out[0] = fmaxf(mi, 0.f);
}

// ---------------------------------------------------------------- launch
extern "C" void kernel_launch(void* const* d_in, const int* in_sizes, int n_in,
                              void* d_out, int out_size, void* d_ws, size_t ws_size,
                              hipStream_t stream) {
  const float* X = (const float*)d_in[0];   // 8192 x 64 f32
  const float* Y = (const float*)d_in[1];   // 8192 x 16 f32
  float* out = (float*)d_out;               // scalar f32

  float* sqx   = (float*)d_ws;              // 8192 f32
  float* sqy   = sqx + NPTS;                // 8192 f32
  int*   kth   = (int*)(sqy + NPTS);        // 8192 i32
  float* accum = (float*)(kth + NPTS);      // 1 f32

  k_norms <<<NPTS / 256, 256, 0, stream>>>(X, Y, sqx, sqy, accum);
  k_select<<<NPTS / 16,  256, 0, stream>>>(X, Y, sqx, sqy, kth);
  k_count <<<NPTS / 16,  256, 0, stream>>>(X, Y, sqx, sqy, kth, accum);
  k_final <<<1, 1, 0, stream>>>(accum, out);
}